// SpatialAttentionLayer_68504728371821
// MI455X (gfx1250) — compile-verified
//
#include <hip/hip_runtime.h>
#include <hip/hip_bf16.h>

#define BB 32
#define NN 1024
#define FF 64
#define TT 64

typedef __attribute__((ext_vector_type(2))) float v2f;
typedef __attribute__((ext_vector_type(8))) float v8f;
typedef int v4i_ __attribute__((vector_size(4 * sizeof(int))));

#if defined(__has_builtin)
#  if __has_builtin(__builtin_amdgcn_global_load_async_to_lds_b128)
#    define HAVE_ASYNC_LDS 1
#  endif
#endif
#ifndef HAVE_ASYNC_LDS
#  define HAVE_ASYNC_LDS 0
#endif

#if HAVE_ASYNC_LDS
#pragma message("gfx1250: using __builtin_amdgcn_global_load_async_to_lds_b128")
#else
#pragma message("gfx1250: async-to-LDS builtin NOT found, using blocking fallback")
#endif

// 16-byte global -> LDS copy (async DMA when available)
__device__ __forceinline__ void copy16_g2l(const float* __restrict__ g, float* l)
{
#if HAVE_ASYNC_LDS
    __builtin_amdgcn_global_load_async_to_lds_b128(
        (__attribute__((address_space(1))) v4i_*)g,
        (__attribute__((address_space(3))) v4i_*)l,
        0, 0);
#else
    *(float4*)l = *(const float4*)g;
#endif
}

// ---------------------------------------------------------------------------
// Kernel 1: fused  lhs0 = x@W1 (over t),  lhs = lhs0@W2,  rhs = W3@x (over f)
// One block per (b,n). x tile 64x64 staged in LDS (stride 65, conflict-free).
// ---------------------------------------------------------------------------
__global__ __launch_bounds__(256) void k1_lhs_rhs(
    const float* __restrict__ x, const float* __restrict__ W1,
    const float* __restrict__ W2, const float* __restrict__ W3,
    float* __restrict__ lhs, float* __restrict__ rhs)
{
    __shared__ float tile[64 * 65];
    __shared__ float W2s[64 * 64];
    __shared__ float W1s[64];
    __shared__ float W3s[64];
    __shared__ float l0[64];

    const int blk = blockIdx.x;                 // b*N + n
    const int tid = threadIdx.x;
    const float* xp = x + (size_t)blk * (FF * TT);

    if (tid < 64)        W1s[tid]      = W1[tid];
    else if (tid < 128)  W3s[tid - 64] = W3[tid - 64];
    else {
        // threads 128..255 preload W2 (1024 float4s, 8 each)
        const int lt = tid - 128;
        #pragma unroll
        for (int i = 0; i < 8; ++i) {
            int q = lt + i * 128;
            ((float4*)W2s)[q] = ((const float4*)W2)[q];
        }
    }

    // stage x tile: 1024 float4s, fully coalesced
    #pragma unroll
    for (int i = 0; i < 4; ++i) {
        int q  = tid + i * 256;                 // float4 index
        float4 v = ((const float4*)xp)[q];
        int f  = q >> 4;
        int t4 = (q & 15) << 2;
        float* d = &tile[f * 65 + t4];
        d[0] = v.x; d[1] = v.y; d[2] = v.z; d[3] = v.w;
    }
    __syncthreads();

    if (tid < 64) {                             // rhs[t] = sum_f W3[f]*x[f][t]
        const int t = tid;
        float acc = 0.f;
        #pragma unroll
        for (int f = 0; f < 64; ++f) acc += W3s[f] * tile[f * 65 + t];
        rhs[(size_t)blk * TT + t] = acc;
    } else if (tid < 128) {                     // lhs0[f] = sum_t x[f][t]*W1[t]
        const int f = tid - 64;
        float acc = 0.f;
        #pragma unroll
        for (int t = 0; t < 64; ++t) acc += tile[f * 65 + t] * W1s[t];
        l0[f] = acc;
    }
    __syncthreads();

    if (tid < 64) {                             // lhs[t] = sum_f lhs0[f]*W2[f][t]
        const int t = tid;
        float acc = 0.f;
        #pragma unroll
        for (int f = 0; f < 64; ++f) acc += l0[f] * W2s[f * 64 + t];
        lhs[(size_t)blk * TT + t] = acc;
    }
}

// ---------------------------------------------------------------------------
// Kernel 2: P[b,n,m] = sigmoid( sum_t lhs[b,n,t]*rhs[b,m,t] + bs[n,m] )
// 64x64 tile per block, 8 waves, each wave owns two 16x16 WMMA accumulators.
// ---------------------------------------------------------------------------
__global__ __launch_bounds__(256) void k2_product_sigmoid(
    const float* __restrict__ lhs, const float* __restrict__ rhs,
    const float* __restrict__ bs, float* __restrict__ P)
{
    __shared__ float Lt[64 * 65];
    __shared__ float Rt[64 * 65];

    const int blk = blockIdx.x;                 // b*256 + tn*16 + tm
    const int b  = blk >> 8;
    const int tn = (blk >> 4) & 15;             // output row tile (n)
    const int tm = blk & 15;                    // output col tile (m)
    const int tid = threadIdx.x;

    const float* Lp = lhs + ((size_t)b * NN + tn * 64) * TT;
    const float* Rp = rhs + ((size_t)b * NN + tm * 64) * TT;

    #pragma unroll
    for (int i = 0; i < 4; ++i) {
        int q  = tid + i * 256;
        int r  = q >> 4;
        int c4 = (q & 15) << 2;
        float4 va = ((const float4*)Lp)[q];
        float* d = &Lt[r * 65 + c4];
        d[0] = va.x; d[1] = va.y; d[2] = va.z; d[3] = va.w;
        float4 vb = ((const float4*)Rp)[q];
        float* e = &Rt[r * 65 + c4];
        e[0] = vb.x; e[1] = vb.y; e[2] = vb.z; e[3] = vb.w;
    }
    __syncthreads();

    const int wave = tid >> 5;
    const int lane = tid & 31;
    const int mt  = wave & 3;                   // row subtile
    const int ntb = (wave >> 2) << 1;           // col subtile base (0 or 2)

    v8f c0 = {}; v8f c1 = {};
    const int lr  = mt * 16 + (lane & 15);      // A: M = lane%16
    const int rr0 = ntb * 16 + (lane & 15);     // B cols come from rhs rows
    const int rr1 = rr0 + 16;
    const int ko  = (lane >> 4) << 1;           // K pair select

    #pragma unroll
    for (int k = 0; k < 64; k += 4) {
        v2f a, b0, b1;
        a.x  = Lt[lr  * 65 + k + ko];  a.y  = Lt[lr  * 65 + k + ko + 1];
        b0.x = Rt[rr0 * 65 + k + ko];  b0.y = Rt[rr0 * 65 + k + ko + 1];
        b1.x = Rt[rr1 * 65 + k + ko];  b1.y = Rt[rr1 * 65 + k + ko + 1];
        c0 = __builtin_amdgcn_wmma_f32_16x16x4_f32(false, a, false, b0, (short)0, c0, false, false);
        c1 = __builtin_amdgcn_wmma_f32_16x16x4_f32(false, a, false, b1, (short)0, c1, false, false);
    }

    // epilogue: +bs, sigmoid, store P
    const size_t off = (size_t)b * NN * NN;
    const int gn_base = tn * 64 + mt * 16 + ((lane >> 4) << 3);
    const int gm0 = tm * 64 + ntb * 16 + (lane & 15);
    #pragma unroll
    for (int r = 0; r < 8; ++r) {
        const int gn = gn_base + r;
        const size_t i0 = (size_t)gn * NN + gm0;
        const size_t i1 = i0 + 16;
        float v0 = c0[r] + bs[i0];
        float v1 = c1[r] + bs[i1];
        v0 = 1.f / (1.f + __expf(-v0));
        v1 = 1.f / (1.f + __expf(-v1));
        P[off + i0] = v0;
        P[off + i1] = v1;
    }
}

// ---------------------------------------------------------------------------
// Kernel 3: S[b] = Vs @ P[b]   (1024x1024x1024 fp32 WMMA GEMM per batch)
// Double-buffered LDS. B tile (P[k][n]) is unpadded stride-64 (column-indexed
// frag reads -> conflict-free) and filled with async b128 global->LDS DMA.
// A tile (Vs[m][k]) needs stride-65 padding (row-indexed frag reads), so it is
// software-pipelined through registers: global loads issued one chunk ahead,
// ds_stores committed after the WMMA block.
// ---------------------------------------------------------------------------
__global__ __launch_bounds__(256) void k3_vs_gemm(
    const float* __restrict__ Vs, const float* __restrict__ P,
    float* __restrict__ S)
{
    __shared__ float At[2][64 * 65];            // Vs tile  [m][k], padded
    __shared__ float Bt[2][64 * 64];            // P  tile  [k][n], linear

    const int blk = blockIdx.x;                 // b*256 + tr*16 + tc
    const int b  = blk >> 8;
    const int tr = (blk >> 4) & 15;
    const int tc = blk & 15;
    const int tid  = threadIdx.x;
    const int wave = tid >> 5;
    const int lane = tid & 31;
    const int mt  = wave & 3;
    const int ntb = (wave >> 2) << 1;

    const float* Pb = P + (size_t)b * NN * NN;

    v8f c0 = {}; v8f c1 = {};
    const int ar  = mt * 16 + (lane & 15);
    const int bc0 = ntb * 16 + (lane & 15);
    const int bc1 = bc0 + 16;
    const int ko  = (lane >> 4) << 1;

    float4 va[4];

    // ---- prologue: stage chunk 0 into buffer 0 ----
    #pragma unroll
    for (int i = 0; i < 4; ++i) {               // async B tile (4 instr/wave)
        int q = tid + (i << 8);
        copy16_g2l(&Pb[(size_t)(q >> 4) * NN + tc * 64 + ((q & 15) << 2)],
                   &Bt[0][q << 2]);
    }
    #pragma unroll
    for (int i = 0; i < 4; ++i) {               // A tile via registers
        int q = tid + (i << 8);
        va[i] = *(const float4*)&Vs[(size_t)(tr * 64 + (q >> 4)) * NN + ((q & 15) << 2)];
    }
    #pragma unroll
    for (int i = 0; i < 4; ++i) {
        int q = tid + (i << 8);
        float* d = &At[0][(q >> 4) * 65 + ((q & 15) << 2)];
        d[0] = va[i].x; d[1] = va[i].y; d[2] = va[i].z; d[3] = va[i].w;
    }

    for (int ic = 0; ic < 16; ++ic) {
        const int buf = ic & 1;
        const int kcn = (ic + 1) << 6;          // next chunk's k offset

        if (ic < 15) {
            // issue next chunk: async B into other buffer, A into registers
            #pragma unroll
            for (int i = 0; i < 4; ++i) {
                int q = tid + (i << 8);
                copy16_g2l(&Pb[(size_t)(kcn + (q >> 4)) * NN + tc * 64 + ((q & 15) << 2)],
                           &Bt[buf ^ 1][q << 2]);
            }
            #pragma unroll
            for (int i = 0; i < 4; ++i) {
                int q = tid + (i << 8);
                va[i] = *(const float4*)&Vs[(size_t)(tr * 64 + (q >> 4)) * NN + kcn + ((q & 15) << 2)];
            }
            asm volatile("s_wait_asynccnt 0x4" ::: "memory");   // current buf done
        } else {
            asm volatile("s_wait_asynccnt 0x0" ::: "memory");
        }
        __syncthreads();                        // staged data visible block-wide

        const float* Atb = At[buf];
        const float* Btb = Bt[buf];
        #pragma unroll
        for (int k = 0; k < 64; k += 4) {
            v2f a, b0, b1;
            a.x  = Atb[ar * 65 + k + ko];       a.y  = Atb[ar * 65 + k + ko + 1];
            b0.x = Btb[(k + ko) * 64 + bc0];    b0.y = Btb[(k + ko + 1) * 64 + bc0];
            b1.x = Btb[(k + ko) * 64 + bc1];    b1.y = Btb[(k + ko + 1) * 64 + bc1];
            c0 = __builtin_amdgcn_wmma_f32_16x16x4_f32(false, a, false, b0, (short)0, c0, false, false);
            c1 = __builtin_amdgcn_wmma_f32_16x16x4_f32(false, a, false, b1, (short)0, c1, false, false);
        }

        if (ic < 15) {
            // commit next A tile (global loads have hidden behind the WMMAs)
            #pragma unroll
            for (int i = 0; i < 4; ++i) {
                int q = tid + (i << 8);
                float* d = &At[buf ^ 1][(q >> 4) * 65 + ((q & 15) << 2)];
                d[0] = va[i].x; d[1] = va[i].y; d[2] = va[i].z; d[3] = va[i].w;
            }
        }
        __syncthreads();                        // drain before buf is refilled
    }

    const size_t off = (size_t)b * NN * NN;
    const int gr_base = tr * 64 + mt * 16 + ((lane >> 4) << 3);
    const int gc0 = tc * 64 + ntb * 16 + (lane & 15);
    #pragma unroll
    for (int r = 0; r < 8; ++r) {
        const int gr = gr_base + r;
        S[off + (size_t)gr * NN + gc0]      = c0[r];
        S[off + (size_t)gr * NN + gc0 + 16] = c1[r];
    }
}

// ---------------------------------------------------------------------------
// Kernel 4: in-place softmax over axis=1 (per (b,m) column over n)
// Block owns 64 columns; 4 row-groups per column, online max/sum + LDS merge.
// ---------------------------------------------------------------------------
__global__ __launch_bounds__(256) void k4_softmax(float* __restrict__ S)
{
    __shared__ float mxs[4][64];
    __shared__ float sms[4][64];

    const int blk = blockIdx.x;                 // b*16 + (m/64)
    const int b  = blk >> 4;
    const int m0 = (blk & 15) << 6;
    const int tid = threadIdx.x;
    const int tm = tid & 63;
    const int tg = tid >> 6;                    // 0..3

    float* Sb = S + (size_t)b * NN * NN + m0 + tm;

    float mx = -__builtin_inff();
    float sm = 0.f;
    for (int n = tg; n < NN; n += 4) {
        float v  = Sb[(size_t)n * NN];
        float nm = fmaxf(mx, v);
        sm = sm * __expf(mx - nm) + __expf(v - nm);
        mx = nm;
    }
    mxs[tg][tm] = mx;
    sms[tg][tm] = sm;
    __syncthreads();

    float M = mxs[0][tm];
    #pragma unroll
    for (int j = 1; j < 4; ++j) M = fmaxf(M, mxs[j][tm]);
    float sum = 0.f;
    #pragma unroll
    for (int j = 0; j < 4; ++j) sum += sms[j][tm] * __expf(mxs[j][tm] - M);
    const float inv = 1.f / sum;

    for (int n = tg; n < NN; n += 4) {
        float v = Sb[(size_t)n * NN];
        Sb[(size_t)n * NN] = __expf(v - M) * inv;
    }
}

// ---------------------------------------------------------------------------
extern "C" void kernel_launch(void* const* d_in, const int* in_sizes, int n_in,
                              void* d_out, int out_size, void* d_ws, size_t ws_size,
                              hipStream_t stream)
{
    const float* x  = (const float*)d_in[0];
    const float* W1 = (const float*)d_in[1];
    const float* W2 = (const float*)d_in[2];
    const float* W3 = (const float*)d_in[3];
    const float* bs = (const float*)d_in[4];
    const float* Vs = (const float*)d_in[5];
    float* out = (float*)d_out;

    char* ws = (char*)d_ws;
    const size_t lr_bytes = (size_t)BB * NN * TT * sizeof(float);   // 8 MB each
    float* lhs = (float*)(ws);
    float* rhs = (float*)(ws + lr_bytes);
    float* P   = (float*)(ws + 2 * lr_bytes);                       // 128 MB

    k1_lhs_rhs<<<BB * NN, 256, 0, stream>>>(x, W1, W2, W3, lhs, rhs);
    k2_product_sigmoid<<<BB * 256, 256, 0, stream>>>(lhs, rhs, bs, P);
    k3_vs_gemm<<<BB * 256, 256, 0, stream>>>(Vs, P, out);
    k4_softmax<<<BB * 16, 256, 0, stream>>>(out);
}